// PinSAGE_83133386981999
// MI455X (gfx1250) — compile-verified
//
#include <hip/hip_runtime.h>
#include <math.h>
#include <float.h>

#define NN 50000
#define EE 300000
#define HH 75
#define TT 5
#define FTT 15
#define GG 256
#define LLAYERS 4
#define THC 375      // T*H
#define PQW 750      // 2*T*H
#define POSTK 975    // 13*H
#define KP 76        // padded K / row stride for WMMA A operands (col 75 == 0)
#define BTPQ_COLS 768  // 12 tiles of 64
#define BTLIN_COLS 80  // 5 tiles of 16
#define EPS_BN 1e-5f
#define EPS_STD 1e-5f

typedef __attribute__((ext_vector_type(2))) float v2f;
typedef __attribute__((ext_vector_type(8))) float v8f;

// ---------------- utility zero kernels (graph-capture safe) ----------------
__global__ void k_zero_f(float* p, int n) {
  int i = blockIdx.x * blockDim.x + threadIdx.x;
  if (i < n) p[i] = 0.0f;
}
__global__ void k_zero_i(int* p, int n) {
  int i = blockIdx.x * blockDim.x + threadIdx.x;
  if (i < n) p[i] = 0;
}

// ---------------- degree histogram scalar ----------------
__global__ void k_avglog(const int* dh, float* avg_log) {
  __shared__ float s1[32], s2[32];
  int t = threadIdx.x;
  float d = (float)dh[t];
  s1[t] = logf((float)t + 1.0f) * d;
  s2[t] = d;
  __syncthreads();
  for (int o = 16; o > 0; o >>= 1) {
    if (t < o) { s1[t] += s1[t + o]; s2[t] += s2[t + o]; }
    __syncthreads();
  }
  if (t == 0) avg_log[0] = s1[0] / s2[0];
}

// ---------------- CSR build ----------------
__global__ void k_count(const int* __restrict__ dst, int* __restrict__ cnt) {
  int e = blockIdx.x * blockDim.x + threadIdx.x;
  if (e < EE) atomicAdd(&cnt[dst[e]], 1);
}

__global__ void k_deg(const int* __restrict__ cnt, const float* __restrict__ avg_log,
                      float* __restrict__ slog, float* __restrict__ islog) {
  int n = blockIdx.x * blockDim.x + threadIdx.x;
  if (n < NN) {
    float deg = fmaxf((float)cnt[n], 1.0f);
    float s = logf(deg + 1.0f) / avg_log[0];
    slog[n] = s;
    islog[n] = 1.0f / s;
  }
}

__global__ __launch_bounds__(1024) void k_scan(const int* __restrict__ cnt, int* __restrict__ row_off) {
  __shared__ int part[1024];
  int t = threadIdx.x;
  const int chunk = (NN + 1023) / 1024;
  int base = t * chunk;
  int s = 0;
  for (int i = 0; i < chunk; i++) { int idx = base + i; if (idx < NN) s += cnt[idx]; }
  part[t] = s;
  __syncthreads();
  for (int off = 1; off < 1024; off <<= 1) {
    int v = 0;
    if (t >= off) v = part[t - off];
    __syncthreads();
    part[t] += v;
    __syncthreads();
  }
  int run = part[t] - s;  // exclusive
  for (int i = 0; i < chunk; i++) {
    int idx = base + i;
    if (idx < NN) { row_off[idx] = run; run += cnt[idx]; }
  }
  if (t == 1023) row_off[NN] = part[1023];
}

__global__ void k_csr_fill(const int* __restrict__ src, const int* __restrict__ dst,
                           const int* __restrict__ row_off, int* __restrict__ cursor,
                           int* __restrict__ csr_src) {
  int e = blockIdx.x * blockDim.x + threadIdx.x;
  if (e < EE) {
    int d = dst[e];
    int pos = row_off[d] + atomicAdd(&cursor[d], 1);
    csr_src[pos] = src[e];
  }
}

// ---------------- node embedding gather (padded stride) ----------------
__global__ void k_embed(const int* __restrict__ x, const float* __restrict__ node_emb,
                        float* __restrict__ h) {
  int i = blockIdx.x * blockDim.x + threadIdx.x;
  if (i < NN * HH) {
    int n = i / HH, c = i % HH;
    h[(size_t)n * KP + c] = node_emb[x[n] * HH + c];
  }
}

// ---------------- per-layer tiny vectors: ea (75), cvec (375) ----------------
__global__ void k_ea(const float* __restrict__ edge_emb, const float* __restrict__ ew,
                     const float* __restrict__ eb, float* __restrict__ ea) {
  int f = threadIdx.x;
  if (f < HH) {
    float a = eb[f];
    for (int d = 0; d < 50; d++) a += edge_emb[d] * ew[d * HH + f];
    ea[f] = a;
  }
}

__global__ void k_cvec(const float* __restrict__ ea, const float* __restrict__ pre_w_l,
                       const float* __restrict__ pre_b_l, float* __restrict__ cvec) {
  int c = blockIdx.x * blockDim.x + threadIdx.x;
  if (c < THC) {
    int t = c / HH, o = c % HH;
    float a = pre_b_l[c];
    const float* w = pre_w_l + (t * 225 + 150) * HH + o;
    for (int f = 0; f < HH; f++) a += ea[f] * w[f * HH];
    cvec[c] = a;
  }
}

// ---------------- weight pack kernels (K-major, zero padded) ----------------
// BT_pq[col*KP + k]:  col<375 -> pre_w[t, k, o] ; col>=375 -> pre_w[t, 75+k, o]
__global__ void k_pack_pq(const float* __restrict__ pre_w_l, float* __restrict__ BT) {
  int i = blockIdx.x * blockDim.x + threadIdx.x;
  if (i < BTPQ_COLS * KP) {
    int col = i / KP, k = i % KP;
    float v = 0.0f;
    if (col < PQW && k < HH) {
      int part = col >= THC;
      int r = col - (part ? THC : 0);
      int t = r / HH, o = r % HH;
      int f = (part ? HH : 0) + k;
      v = pre_w_l[(t * 225 + f) * HH + o];
    }
    BT[i] = v;
  }
}

__global__ void k_pack_lin(const float* __restrict__ lin_w_l, float* __restrict__ BT) {
  int i = blockIdx.x * blockDim.x + threadIdx.x;
  if (i < BTLIN_COLS * KP) {
    int col = i / KP, k = i % KP;
    BT[i] = (col < HH && k < HH) ? lin_w_l[k * HH + col] : 0.0f;
  }
}

// ---------------- WMMA GEMM: PQ = h(N x 75, stride KP) @ Wcat(75 x 750) ----------------
// 4 waves / block, each wave computes a 16x64 tile (4 accumulators).
__global__ __launch_bounds__(128) void k_pq(const float* __restrict__ hp,
                                            const float* __restrict__ BT,
                                            float* __restrict__ PQ) {
  int w = threadIdx.x >> 5;
  int lane = threadIdx.x & 31;
  int mt = blockIdx.x * 4 + w;
  if (mt >= NN / 16) return;
  int half = lane >> 4, q = lane & 15;
  int mrow = mt * 16 + q;
  int nbase = blockIdx.y * 64;
  const float* arow = hp + (size_t)mrow * KP;
  const float* bp[4];
#pragma unroll
  for (int j = 0; j < 4; j++) bp[j] = BT + (size_t)(nbase + j * 16 + q) * KP;
  v8f accs[4];
#pragma unroll
  for (int j = 0; j < 4; j++) accs[j] = (v8f){};
  for (int k0 = 0; k0 < HH; k0 += 4) {
    int ka = k0 + 2 * half;
    v2f a = *(const v2f*)(arow + ka);
#pragma unroll
    for (int j = 0; j < 4; j++) {
      v2f b = *(const v2f*)(bp[j] + ka);
      accs[j] = __builtin_amdgcn_wmma_f32_16x16x4_f32(false, a, false, b, (short)0, accs[j], false, false);
    }
  }
#pragma unroll
  for (int j = 0; j < 4; j++) {
    int ncol = nbase + j * 16 + q;
    if (ncol < PQW) {
#pragma unroll
      for (int i = 0; i < 8; i++) {
        PQ[(size_t)(mt * 16 + i + 8 * half) * PQW + ncol] = accs[j][i];
      }
    }
  }
}

// ---------------- WMMA GEMM: C = A(N x 75, stride KP) @ W(75 x 75) + bias ----------------
// 4 waves / block, each wave computes 16 x 80 (5 accumulators -> all cols).
__global__ __launch_bounds__(128) void k_lin(const float* __restrict__ A,
                                             const float* __restrict__ BT,
                                             const float* __restrict__ bias,
                                             float* __restrict__ C) {
  int w = threadIdx.x >> 5;
  int lane = threadIdx.x & 31;
  int mt = blockIdx.x * 4 + w;
  if (mt >= NN / 16) return;
  int half = lane >> 4, q = lane & 15;
  int mrow = mt * 16 + q;
  const float* arow = A + (size_t)mrow * KP;
  const float* bp[5];
#pragma unroll
  for (int j = 0; j < 5; j++) bp[j] = BT + (size_t)(j * 16 + q) * KP;
  v8f accs[5];
#pragma unroll
  for (int j = 0; j < 5; j++) accs[j] = (v8f){};
  for (int k0 = 0; k0 < HH; k0 += 4) {
    int ka = k0 + 2 * half;
    v2f a = *(const v2f*)(arow + ka);
#pragma unroll
    for (int j = 0; j < 5; j++) {
      v2f b = *(const v2f*)(bp[j] + ka);
      accs[j] = __builtin_amdgcn_wmma_f32_16x16x4_f32(false, a, false, b, (short)0, accs[j], false, false);
    }
  }
#pragma unroll
  for (int j = 0; j < 5; j++) {
    int ncol = j * 16 + q;
    if (ncol < HH) {
      float bb = bias[ncol];
#pragma unroll
      for (int i = 0; i < 8; i++) {
        C[(size_t)(mt * 16 + i + 8 * half) * KP + ncol] = accs[j][i] + bb;
      }
    }
  }
}

// ---------------- fused aggregation + post-tower matmul ----------------
// msg[e] = P[dst] + Q[src] + cvec  =>  stats over Q shifted by a = P[n]+cvec
// y[n,t,o] = post_b + sum_f h*W + sum_j stat_j*(W1 + slog*W2 + islog*W3)
#define NB 8
__global__ __launch_bounds__(256) void k_aggpost(
    const float* __restrict__ PQ, const float* __restrict__ cvec,
    const int* __restrict__ row_off, const int* __restrict__ csr_src,
    const float* __restrict__ h, const float* __restrict__ slog,
    const float* __restrict__ islog, const float* __restrict__ post_w_l,
    const float* __restrict__ post_b_l, float* __restrict__ ypre) {
  __shared__ float s_stat[NB][4][THC + 1];   // mean,min,max,std  (~48 KB)
  __shared__ float s_h[NB][HH + 1];
  int n0 = blockIdx.x * NB;
  int tid = threadIdx.x;
  for (int idx = tid; idx < NB * HH; idx += 256) {
    int i = idx / HH, f = idx % HH;
    int n = n0 + i;
    s_h[i][f] = (n < NN) ? h[(size_t)n * KP + f] : 0.0f;
  }
  for (int idx = tid; idx < NB * THC; idx += 256) {
    int i = idx / THC, c = idx % THC;
    int n = n0 + i;
    if (n < NN) {
      int e0 = row_off[n], e1 = row_off[n + 1];
      float s1 = 0.0f, s2 = 0.0f, mn = FLT_MAX, mx = -FLT_MAX;
      for (int e = e0; e < e1; e++) {
        int s = csr_src[e];
        float qv = PQ[(size_t)s * PQW + THC + c];
        s1 += qv; s2 += qv * qv;
        mn = fminf(mn, qv); mx = fmaxf(mx, qv);
      }
      int deg = e1 - e0;
      float mean, stdv, mnv, mxv;
      if (deg == 0) {
        mean = 0.0f; stdv = sqrtf(EPS_STD); mnv = 0.0f; mxv = 0.0f;
      } else {
        float a = PQ[(size_t)n * PQW + c] + cvec[c];
        float id = 1.0f / (float)deg;
        float m1 = s1 * id;
        mean = a + m1;
        stdv = sqrtf(fmaxf(s2 * id - m1 * m1, 0.0f) + EPS_STD);
        mnv = a + mn; mxv = a + mx;
      }
      s_stat[i][0][c] = mean;
      s_stat[i][1][c] = mnv;
      s_stat[i][2][c] = mxv;
      s_stat[i][3][c] = stdv;
    }
  }
  __syncthreads();
  for (int out = tid; out < NB * HH; out += 256) {
    int i = out / HH, r = out % HH;
    int n = n0 + i;
    if (n >= NN) continue;
    int t = r / FTT, o = r % FTT;
    const float* pw = post_w_l + (t * POSTK) * FTT + o;   // pw[f*FTT] = W[t][f][o]
    float acc = post_b_l[t * FTT + o];
    for (int f = 0; f < HH; f++) acc += s_h[i][f] * pw[f * FTT];
    float sl = slog[n], isl = islog[n];
    int tbase = t * HH;
    for (int j = 0; j < 300; j++) {
      int g = j / HH, hh2 = j % HH;
      float st = s_stat[i][g][tbase + hh2];
      float wv = pw[(HH + j) * FTT] + sl * pw[(THC + j) * FTT] + isl * pw[(675 + j) * FTT];
      acc += st * wv;
    }
    ypre[(size_t)n * KP + r] = acc;
  }
}

// ---------------- batch-norm statistics ----------------
__global__ __launch_bounds__(256) void k_bnstat(const float* __restrict__ y,
                                                float* __restrict__ bsum,
                                                float* __restrict__ bsq) {
  __shared__ float ss[HH], sq[HH];
  for (int c = threadIdx.x; c < HH; c += 256) { ss[c] = 0.0f; sq[c] = 0.0f; }
  __syncthreads();
  int n = blockIdx.x * 256 + threadIdx.x;
  int lane = threadIdx.x & 31;
  for (int c = 0; c < HH; c++) {
    float v = (n < NN) ? y[(size_t)n * KP + c] : 0.0f;
    float v2 = v * v;
    for (int off = 16; off > 0; off >>= 1) {
      v += __shfl_down(v, off, 32);
      v2 += __shfl_down(v2, off, 32);
    }
    if (lane == 0) { atomicAdd(&ss[c], v); atomicAdd(&sq[c], v2); }
  }
  __syncthreads();
  for (int c = threadIdx.x; c < HH; c += 256) {
    atomicAdd(&bsum[c], ss[c]);
    atomicAdd(&bsq[c], sq[c]);
  }
}

__global__ void k_bnapply(float* __restrict__ y, const float* __restrict__ bsum,
                          const float* __restrict__ bsq, const float* __restrict__ g,
                          const float* __restrict__ b) {
  int i = blockIdx.x * blockDim.x + threadIdx.x;
  if (i < NN * HH) {
    int n = i / HH, c = i % HH;
    float mu = bsum[c] * (1.0f / NN);
    float var = bsq[c] * (1.0f / NN) - mu * mu;
    size_t a = (size_t)n * KP + c;
    float v = (y[a] - mu) * rsqrtf(var + EPS_BN) * g[c] + b[c];
    y[a] = fmaxf(v, 0.0f);
  }
}

// ---------------- graph pooling + MLP ----------------
__global__ void k_pool(const float* __restrict__ h, const int* __restrict__ batch,
                       float* __restrict__ gpool) {
  int i = blockIdx.x * blockDim.x + threadIdx.x;
  if (i < NN * HH) {
    int n = i / HH, c = i % HH;
    atomicAdd(&gpool[batch[n] * HH + c], h[(size_t)n * KP + c]);
  }
}

__global__ void k_mlp1(const float* __restrict__ gp, const float* __restrict__ w,
                       const float* __restrict__ b, float* __restrict__ out) {
  int i = blockIdx.x * blockDim.x + threadIdx.x;
  if (i < GG * 50) {
    int g = i / 50, o = i % 50;
    float a = b[o];
    for (int f = 0; f < HH; f++) a += gp[g * HH + f] * w[f * 50 + o];
    out[i] = fmaxf(a, 0.0f);
  }
}
__global__ void k_mlp2(const float* __restrict__ m1, const float* __restrict__ w,
                       const float* __restrict__ b, float* __restrict__ out) {
  int i = blockIdx.x * blockDim.x + threadIdx.x;
  if (i < GG * 25) {
    int g = i / 25, o = i % 25;
    float a = b[o];
    for (int f = 0; f < 50; f++) a += m1[g * 50 + f] * w[f * 25 + o];
    out[i] = fmaxf(a, 0.0f);
  }
}
__global__ void k_mlp3(const float* __restrict__ m2, const float* __restrict__ w,
                       const float* __restrict__ b, float* __restrict__ out) {
  int i = blockIdx.x * blockDim.x + threadIdx.x;
  if (i < GG * 3) {
    int g = i / 3, o = i % 3;
    float a = b[o];
    for (int f = 0; f < 25; f++) a += m2[g * 25 + f] * w[f * 3 + o];
    out[i] = a;
  }
}

// ---------------- host launcher ----------------
extern "C" void kernel_launch(void* const* d_in, const int* in_sizes, int n_in,
                              void* d_out, int out_size, void* d_ws, size_t ws_size,
                              hipStream_t stream) {
  const int* x          = (const int*)d_in[0];
  const int* ei         = (const int*)d_in[1];
  const int* src        = ei;
  const int* dst        = ei + EE;
  const int* batch      = (const int*)d_in[3];
  const int* deg_hist   = (const int*)d_in[4];
  const float* node_emb = (const float*)d_in[5];
  const float* edge_emb = (const float*)d_in[6];
  const float* eenc_w   = (const float*)d_in[7];
  const float* eenc_b   = (const float*)d_in[8];
  const float* pre_w    = (const float*)d_in[9];
  const float* pre_b    = (const float*)d_in[10];
  const float* post_w   = (const float*)d_in[11];
  const float* post_b   = (const float*)d_in[12];
  const float* lin_w    = (const float*)d_in[13];
  const float* lin_b    = (const float*)d_in[14];
  const float* bn_g     = (const float*)d_in[15];
  const float* bn_b     = (const float*)d_in[16];
  const float* mlp_w1   = (const float*)d_in[17];
  const float* mlp_b1   = (const float*)d_in[18];
  const float* mlp_w2   = (const float*)d_in[19];
  const float* mlp_b2   = (const float*)d_in[20];
  const float* mlp_w3   = (const float*)d_in[21];
  const float* mlp_b3   = (const float*)d_in[22];

  // workspace carve-up
  char* wsp = (char*)d_ws;
  size_t off = 0;
  auto alloc = [&](size_t bytes) -> void* {
    void* p = wsp + off;
    off = (off + bytes + 255) & ~((size_t)255);
    return p;
  };
  float* avg_log = (float*)alloc(64);
  int*   cnt     = (int*)alloc(sizeof(int) * NN);
  int*   row_off = (int*)alloc(sizeof(int) * (NN + 1));
  int*   cursor  = (int*)alloc(sizeof(int) * NN);
  int*   csr_src = (int*)alloc(sizeof(int) * EE);
  float* slog    = (float*)alloc(sizeof(float) * NN);
  float* islog   = (float*)alloc(sizeof(float) * NN);
  float* ea      = (float*)alloc(sizeof(float) * 128);
  float* cvec    = (float*)alloc(sizeof(float) * 512);
  float* bnsum   = (float*)alloc(sizeof(float) * 128);
  float* bnsq    = (float*)alloc(sizeof(float) * 128);
  float* gpool   = (float*)alloc(sizeof(float) * GG * HH);
  float* m1      = (float*)alloc(sizeof(float) * GG * 50);
  float* m2      = (float*)alloc(sizeof(float) * GG * 25);
  float* BT_pq   = (float*)alloc(sizeof(float) * BTPQ_COLS * KP);
  float* BT_lin  = (float*)alloc(sizeof(float) * BTLIN_COLS * KP);
  float* hA      = (float*)alloc(sizeof(float) * (size_t)NN * KP);
  float* hB      = (float*)alloc(sizeof(float) * (size_t)NN * KP);
  float* ypre    = (float*)alloc(sizeof(float) * (size_t)NN * KP);
  float* PQ      = (float*)alloc(sizeof(float) * (size_t)NN * PQW);
  (void)ws_size; (void)n_in; (void)in_sizes; (void)out_size;

  const int B = 256;
  // setup: zero pads + counters
  k_zero_f<<<((int)(NN * KP) + B - 1) / B, B, 0, stream>>>(hA, NN * KP);
  k_zero_f<<<((int)(NN * KP) + B - 1) / B, B, 0, stream>>>(hB, NN * KP);
  k_zero_f<<<((int)(NN * KP) + B - 1) / B, B, 0, stream>>>(ypre, NN * KP);
  k_zero_i<<<(NN + B - 1) / B, B, 0, stream>>>(cnt, NN);
  k_zero_i<<<(NN + B - 1) / B, B, 0, stream>>>(cursor, NN);
  k_avglog<<<1, 32, 0, stream>>>(deg_hist, avg_log);
  k_count<<<(EE + B - 1) / B, B, 0, stream>>>(dst, cnt);
  k_deg<<<(NN + B - 1) / B, B, 0, stream>>>(cnt, avg_log, slog, islog);
  k_scan<<<1, 1024, 0, stream>>>(cnt, row_off);
  k_csr_fill<<<(EE + B - 1) / B, B, 0, stream>>>(src, dst, row_off, cursor, csr_src);
  k_embed<<<(NN * HH + B - 1) / B, B, 0, stream>>>(x, node_emb, hA);

  float* h_in = hA;
  float* h_nx = hB;
  for (int l = 0; l < LLAYERS; l++) {
    const float* pre_w_l  = pre_w + (size_t)l * TT * 225 * HH;
    const float* pre_b_l  = pre_b + (size_t)l * THC;
    const float* post_w_l = post_w + (size_t)l * TT * POSTK * FTT;
    const float* post_b_l = post_b + (size_t)l * TT * FTT;
    const float* eenc_w_l = eenc_w + (size_t)l * 50 * HH;
    const float* eenc_b_l = eenc_b + (size_t)l * HH;
    const float* lin_w_l  = lin_w + (size_t)l * HH * HH;
    const float* lin_b_l  = lin_b + (size_t)l * HH;
    const float* bn_g_l   = bn_g + (size_t)l * HH;
    const float* bn_b_l   = bn_b + (size_t)l * HH;

    k_ea<<<1, 128, 0, stream>>>(edge_emb, eenc_w_l, eenc_b_l, ea);
    k_cvec<<<2, 256, 0, stream>>>(ea, pre_w_l, pre_b_l, cvec);
    k_pack_pq<<<(BTPQ_COLS * KP + B - 1) / B, B, 0, stream>>>(pre_w_l, BT_pq);
    k_pack_lin<<<(BTLIN_COLS * KP + B - 1) / B, B, 0, stream>>>(lin_w_l, BT_lin);

    dim3 gpq((NN / 16 + 3) / 4, BTPQ_COLS / 64);
    k_pq<<<gpq, 128, 0, stream>>>(h_in, BT_pq, PQ);

    k_aggpost<<<(NN + NB - 1) / NB, 256, 0, stream>>>(
        PQ, cvec, row_off, csr_src, h_in, slog, islog, post_w_l, post_b_l, ypre);

    dim3 glin((NN / 16 + 3) / 4);
    k_lin<<<glin, 128, 0, stream>>>(ypre, BT_lin, lin_b_l, h_nx);

    k_zero_f<<<1, 256, 0, stream>>>(bnsum, 128);
    k_zero_f<<<1, 256, 0, stream>>>(bnsq, 128);
    k_bnstat<<<(NN + 255) / 256, 256, 0, stream>>>(h_nx, bnsum, bnsq);
    k_bnapply<<<(NN * HH + B - 1) / B, B, 0, stream>>>(h_nx, bnsum, bnsq, bn_g_l, bn_b_l);

    float* t = h_in; h_in = h_nx; h_nx = t;
  }

  k_zero_f<<<(GG * HH + B - 1) / B, B, 0, stream>>>(gpool, GG * HH);
  k_pool<<<(NN * HH + B - 1) / B, B, 0, stream>>>(h_in, batch, gpool);
  k_mlp1<<<(GG * 50 + B - 1) / B, B, 0, stream>>>(gpool, mlp_w1, mlp_b1, m1);
  k_mlp2<<<(GG * 25 + B - 1) / B, B, 0, stream>>>(m1, mlp_w2, mlp_b2, m2);
  k_mlp3<<<(GG * 3 + B - 1) / B, B, 0, stream>>>(m2, mlp_w3, mlp_b3, (float*)d_out);
}